// AUX_Model_80272938762938
// MI455X (gfx1250) — compile-verified
//
#include <hip/hip_runtime.h>
#include <hip/hip_bf16.h>

// ---------------------------------------------------------------------------
// Bidirectional 2-layer LSTM (H=6) + FC head on MI455X (gfx1250, wave32).
//
// One wave = 16 batch samples. Gate pre-activations computed with
// v_wmma_f32_16x16x4_f32:
//   D(16x16) = A(16x4) x B(4x16) + C
//   A = weight tile [gates x K-chunk], shared across the wave (loaded once)
//   B = per-sample inputs [K-chunk x 16 samples]
//   C = bias (initial accumulator), chained across K chunks
// Gate rows padded to 8 per gate: rows [i*8 | f*8 | g*8 | o*8] -> two M tiles.
// D layout (32-bit C/D, 16x16): lane n (<16) vgpr r = row r; lane n+16 = row 8+r.
//   => tile0: lanes0-15 hold i-gates, lanes16-31 hold f-gates (same VGPRs)
//      tile1: lanes0-15 hold g-gates, lanes16-31 hold o-gates
// Pointwise LSTM update then needs only shfl_xor(...,16) lane swaps.
//
// A-matrix 16x4 f32 layout (ISA 7.12.2): lanes0-15: M=lane, v0=K0,v1=K1;
// lanes16-31: M=lane-16, v0=K2,v1=K3.  B assumed mirrored (lane=N col,
// low lanes K0/K1, high lanes K2/K3).
//
// All B-operand construction is branchless: loads use arithmetic per-lane
// addressing, register/memory mixes use v_cndmask selects, so EXEC stays
// all-ones through the whole serial loop (required for WMMA, and avoids
// save/restore-exec branches on the latency-critical path).
// ---------------------------------------------------------------------------

typedef __attribute__((ext_vector_type(2))) float v2f;
typedef __attribute__((ext_vector_type(8))) float v8f;

#define LSTM_H 6

static __device__ __forceinline__ float sigm(float x) {
  return __builtin_amdgcn_rcpf(1.0f + __expf(-x));
}
static __device__ __forceinline__ float tanh_fast(float x) {
  return 1.0f - 2.0f * __builtin_amdgcn_rcpf(__expf(2.0f * x) + 1.0f);
}
static __device__ __forceinline__ float sx16(float v) {
  return __shfl_xor(v, 16, 32);   // lane n <-> lane n^16 (wave32)
}
static __device__ __forceinline__ v8f wmma4(v2f a, v2f b, v8f c) {
  // (neg_a, A, neg_b, B, c_mod, C, reuse_a, reuse_b)
  return __builtin_amdgcn_wmma_f32_16x16x4_f32(false, a, false, b, (short)0, c,
                                               false, false);
}

// padded gate row m in [0,32): groups of 8 = [i, f, g, o], 6 valid rows each
static __device__ __forceinline__ int gate_row(int m) {
  int idx = m & 7;
  return (idx < LSTM_H) ? ((m >> 3) * LSTM_H + idx) : -1;
}

// combined weight W[m][k] with K = [x(0..D-1) ; h(D..D+5)], zero padding
static __device__ __forceinline__ float wcomb(const float* Wih, const float* Whh,
                                              int D, int m, int k) {
  int r = gate_row(m);
  if (r < 0) return 0.0f;
  if (k < D) return Wih[r * D + k];
  k -= D;
  return (k < LSTM_H) ? Whh[r * LSTM_H + k] : 0.0f;
}

// Load A tiles (2 M-tiles x nch K-chunks) + bias accumulators for this wave.
static __device__ __forceinline__ void load_A(const float* Wih, const float* Whh,
                                              const float* bias, int D, int nch,
                                              v2f a[2][5], v8f cini[2],
                                              int lm, int half) {
  for (int t = 0; t < 2; ++t) {
    int m = t * 16 + lm;                    // A rows 0..15 of this tile
#pragma unroll
    for (int c = 0; c < 5; ++c) {
      if (c < nch) {
        int k0 = 4 * c + 2 * half;          // low lanes K0/K1, high lanes K2/K3
        v2f av;
        av.x = wcomb(Wih, Whh, D, m, k0);
        av.y = wcomb(Wih, Whh, D, m, k0 + 1);
        a[t][c] = av;
      }
    }
    v8f ci;
#pragma unroll
    for (int r = 0; r < 8; ++r) {           // D row = (half ? 8+r : r)
      int gr = gate_row(t * 16 + half * 8 + r);
      ci[r] = (gr >= 0) ? bias[gr] : 0.0f;
    }
    cini[t] = ci;
  }
}

static __device__ __forceinline__ void mm(const v2f a[2][5], const v8f cini[2],
                                          const v2f bch[5], int nch,
                                          v8f& A0, v8f& A1) {
  A0 = cini[0];
  A1 = cini[1];
#pragma unroll
  for (int c = 0; c < 5; ++c) {
    if (c < nch) {
      A0 = wmma4(a[0][c], bch[c], A0);
      A1 = wmma4(a[1][c], bch[c], A1);
    }
  }
}

// A0: lanes<16 = i-gate preact, lanes>=16 = f-gate. A1: g / o.
// c[],h[] meaningful in lanes 0-15 (one sample per low lane).
static __device__ __forceinline__ void lstm_pointwise(const v8f& A0, const v8f& A1,
                                                      bool low, float c[6], float h[6]) {
#pragma unroll
  for (int r = 0; r < LSTM_H; ++r) {
    float s_if = sigm(A0[r]);            // low: sig(i), high: sig(f)
    float tg = tanh_fast(A1[r]);
    float so = sigm(A1[r]);
    float go = low ? tg : so;            // low: tanh(g), high: sig(o)
    float fsw = sx16(s_if);              // low lanes receive sig(f)
    float cn = fsw * c[r] + s_if * go;   // c' = sig(f)*c + sig(i)*tanh(g)
    c[r] = cn;
    float osw = sx16(go);                // low lanes receive sig(o)
    h[r] = osw * tanh_fast(cn);          // h' = sig(o)*tanh(c')
  }
}

// Layer-0 B operand: K = [x0..x5, h0..h5]; chunk c holds k=4c..4c+3.
// Branchless: arithmetic addressing for loads, cndmask for reg/mem mixes.
static __device__ __forceinline__ void b_l0(const float* xp, const float h[6],
                                            int half, v2f bch[5]) {
  v2f xlo = *(const v2f*)(xp + 2 * half);   // low: x0,x1 ; high: x2,x3  (b64)
  v2f x45 = *(const v2f*)(xp + 4);          // all lanes: x4,x5          (b64)
  float h0s = sx16(h[0]), h1s = sx16(h[1]), h4s = sx16(h[4]), h5s = sx16(h[5]);
  v2f b1, b2;
  b1.x = half ? h0s : x45.x; b1.y = half ? h1s : x45.y;
  b2.x = half ? h4s : h[2];  b2.y = half ? h5s : h[3];
  bch[0] = xlo; bch[1] = b1; bch[2] = b2;
}

// Layer-1 B operand: K = [f0(6), r0(6), h1(6), 0, 0]  (K padded 18 -> 20).
static __device__ __forceinline__ void b_l1(const float f0[6], const float* r0p,
                                            const float h[6], int half, v2f bch[5]) {
  v2f r01 = *(const v2f*)(r0p);                  // all lanes: r0,r1      (b64)
  v2f rhi = *(const v2f*)(r0p + 2 + 2 * half);   // low: r2,r3; high: r4,r5
  float f2s = sx16(f0[2]), f3s = sx16(f0[3]);
  float h2s = sx16(h[2]),  h3s = sx16(h[3]);
  v2f b0, b1, b3, b4;
  b0.x = half ? f2s : f0[0];    b0.y = half ? f3s : f0[1];
  b1.x = half ? r01.x : f0[4];  b1.y = half ? r01.y : f0[5];
  b3.x = half ? h2s : h[0];     b3.y = half ? h3s : h[1];
  b4.x = half ? 0.0f : h[4];    b4.y = half ? 0.0f : h[5];
  bch[0] = b0; bch[1] = b1; bch[2] = rhi; bch[3] = b3; bch[4] = b4;
}

// ---------------- Kernel 1: layer-0 reverse scan, store r0[t] -------------
__global__ void lstm_l0_bwd(const float* __restrict__ tf, const float* __restrict__ h0,
                            const float* __restrict__ c0,
                            const float* __restrict__ Wih, const float* __restrict__ Whh,
                            const float* __restrict__ bias,
                            float* __restrict__ r0ws, int B, int T) {
  int tid = blockIdx.x * blockDim.x + threadIdx.x;
  int lane = tid & 31, lm = lane & 15, half = lane >> 4;
  int s = (tid >> 5) * 16 + lm;
  if (s >= B) return;

  v2f a[2][5]; v8f ci[2];
  load_A(Wih, Whh, bias, LSTM_H, 3, a, ci, lm, half);

  float h[6], c[6];
  size_t d1 = (size_t)B * LSTM_H;                 // direction 1 (layer0 bwd)
#pragma unroll
  for (int j = 0; j < 6; ++j) {
    h[j] = h0[d1 + (size_t)s * 6 + j];
    c[j] = c0[d1 + (size_t)s * 6 + j];
  }

  for (int t = T - 1; t >= 0; --t) {
    const float* xp = tf + ((size_t)s * T + t) * 6;
    if (t > 0) __builtin_prefetch(xp - 6, 0, 1);  // global_prefetch_b8
    v2f bch[5]; b_l0(xp, h, half, bch);
    v8f A0, A1; mm(a, ci, bch, 3, A0, A1);
    lstm_pointwise(A0, A1, half == 0, c, h);
    if (half == 0) {                               // re-converges before next wmma
      float* wp = r0ws + ((size_t)t * B + s) * 6;  // 8B-aligned (24B records)
      v2f w01; w01.x = h[0]; w01.y = h[1];
      v2f w23; w23.x = h[2]; w23.y = h[3];
      v2f w45; w45.x = h[4]; w45.y = h[5];
      *(v2f*)(wp) = w01; *(v2f*)(wp + 2) = w23; *(v2f*)(wp + 4) = w45;
    }
  }
}

// ------- Kernel 2: fused layer0-fwd + layer1-fwd, layer1-bwd step, FC -----
__global__ void lstm_fused(const float* __restrict__ tf, const float* __restrict__ h0,
                           const float* __restrict__ c0,
                           const float* __restrict__ Wih0f, const float* __restrict__ Whh0f,
                           const float* __restrict__ b0f,
                           const float* __restrict__ Wih1f, const float* __restrict__ Whh1f,
                           const float* __restrict__ b1f,
                           const float* __restrict__ Wih1b, const float* __restrict__ Whh1b,
                           const float* __restrict__ b1b,
                           const float* __restrict__ fcw, const float* __restrict__ fcb,
                           const float* __restrict__ r0ws, float* __restrict__ out,
                           int B, int T) {
  int tid = blockIdx.x * blockDim.x + threadIdx.x;
  int lane = tid & 31, lm = lane & 15, half = lane >> 4;
  int s = (tid >> 5) * 16 + lm;
  if (s >= B) return;

  v2f a0[2][5]; v8f ci0[2]; load_A(Wih0f, Whh0f, b0f, LSTM_H, 3, a0, ci0, lm, half);
  v2f a1[2][5]; v8f ci1[2]; load_A(Wih1f, Whh1f, b1f, 2 * LSTM_H, 5, a1, ci1, lm, half);

  float hA[6], cA[6], hB[6], cB[6];
  size_t d0 = 0, d2 = 2 * (size_t)B * LSTM_H;
#pragma unroll
  for (int j = 0; j < 6; ++j) {
    hA[j] = h0[d0 + (size_t)s * 6 + j];  cA[j] = c0[d0 + (size_t)s * 6 + j];
    hB[j] = h0[d2 + (size_t)s * 6 + j];  cB[j] = c0[d2 + (size_t)s * 6 + j];
  }

  for (int t = 0; t < T; ++t) {
    const float* xp = tf + ((size_t)s * T + t) * 6;
    if (t + 1 < T) __builtin_prefetch(xp + 6, 0, 1);
    // layer 0 forward step -> hA = f0[t]
    v2f bch[5]; b_l0(xp, hA, half, bch);
    v8f A0, A1; mm(a0, ci0, bch, 3, A0, A1);
    lstm_pointwise(A0, A1, half == 0, cA, hA);
    // layer 1 forward step on [f0[t], r0[t], hB]
    const float* r0p = r0ws + ((size_t)t * B + s) * 6;
    b_l1(hA, r0p, hB, half, bch);
    mm(a1, ci1, bch, 5, A0, A1);
    lstm_pointwise(A0, A1, half == 0, cB, hB);
  }
  // hB = f1[T-1]. Layer-1 reverse direction: only its t=T-1 output is needed,
  // which is ONE step from (h0[3], c0[3]) on input l0[T-1] = [f0[T-1], r0[T-1]].
  v2f a2[2][5]; v8f ci2[2]; load_A(Wih1b, Whh1b, b1b, 2 * LSTM_H, 5, a2, ci2, lm, half);
  float hR[6], cR[6];
  size_t d3 = 3 * (size_t)B * LSTM_H;
#pragma unroll
  for (int j = 0; j < 6; ++j) {
    hR[j] = h0[d3 + (size_t)s * 6 + j];  cR[j] = c0[d3 + (size_t)s * 6 + j];
  }
  {
    const float* r0p = r0ws + ((size_t)(T - 1) * B + s) * 6;
    v2f bch[5]; b_l1(hA, r0p, hR, half, bch);
    v8f A0, A1; mm(a2, ci2, bch, 5, A0, A1);
    lstm_pointwise(A0, A1, half == 0, cR, hR);
  }
  // FC head: out = relu([f1_last, r1_first]) @ fcw.T + fcb   (lanes 0-15 only)
  if (half == 0) {
    float v[12];
#pragma unroll
    for (int k = 0; k < 6; ++k) {
      v[k]     = hB[k] > 0.0f ? hB[k] : 0.0f;
      v[6 + k] = hR[k] > 0.0f ? hR[k] : 0.0f;
    }
#pragma unroll
    for (int j = 0; j < 6; ++j) {
      float acc = fcb[j];
#pragma unroll
      for (int k = 0; k < 12; ++k) acc = fmaf(fcw[j * 12 + k], v[k], acc);
      out[(size_t)s * 6 + j] = acc;
    }
  }
}

// ---------------------------------------------------------------------------
extern "C" void kernel_launch(void* const* d_in, const int* in_sizes, int n_in,
                              void* d_out, int out_size, void* d_ws, size_t ws_size,
                              hipStream_t stream) {
  const float* tf    = (const float*)d_in[0];
  const float* h0    = (const float*)d_in[1];
  const float* c0    = (const float*)d_in[2];
  const float* Wih0f = (const float*)d_in[3];
  const float* Whh0f = (const float*)d_in[4];
  const float* b0f   = (const float*)d_in[5];
  const float* Wih0b = (const float*)d_in[6];
  const float* Whh0b = (const float*)d_in[7];
  const float* b0b   = (const float*)d_in[8];
  const float* Wih1f = (const float*)d_in[9];
  const float* Whh1f = (const float*)d_in[10];
  const float* b1f   = (const float*)d_in[11];
  const float* Wih1b = (const float*)d_in[12];
  const float* Whh1b = (const float*)d_in[13];
  const float* b1b   = (const float*)d_in[14];
  const float* fcw   = (const float*)d_in[15];
  const float* fcb   = (const float*)d_in[16];

  int B = in_sizes[1] / (4 * LSTM_H);          // h0 is (4, B, H)
  int T = in_sizes[0] / (B * LSTM_H);          // time_feature is (B, T, 6)

  size_t need = (size_t)B * T * LSTM_H * sizeof(float);  // r0 buffer, ~100 MB
  if (ws_size < need) return;
  float* r0ws = (float*)d_ws;

  int waves = (B + 15) / 16;
  int threads = waves * 32;
  int block = 256;
  int grid = (threads + block - 1) / block;

  lstm_l0_bwd<<<grid, block, 0, stream>>>(tf, h0, c0, Wih0b, Whh0b, b0b, r0ws, B, T);
  lstm_fused<<<grid, block, 0, stream>>>(tf, h0, c0, Wih0f, Whh0f, b0f,
                                         Wih1f, Whh1f, b1f, Wih1b, Whh1b, b1b,
                                         fcw, fcb, r0ws, (float*)d_out, B, T);
}